// BTNetEuropean_79242146611469
// MI455X (gfx1250) — compile-verified
//
#include <hip/hip_runtime.h>

// CDNA5 / gfx1250, wave32.
typedef __attribute__((ext_vector_type(2))) float v2f;   // A/B operands of V_WMMA_F32_16X16X4_F32
typedef __attribute__((ext_vector_type(8))) float v8f;   // 16x16 f32 C/D accumulator

#define N_DIM 1024
#define NP1   1025          // N_DIM + 1
#define KPAD  1028          // NP1 rounded up to multiple of 4 (WMMA K-step)

// ---------------------------------------------------------------------------
// Phase 1: coefficient vector c = e0^T * A^N via the same stable recurrence
// the reference scan applies ( c'[j] = w0*c[j] + w1*c[j-1] ), run N_DIM times
// on a single 1025-vector in LDS. Also emits zero-padded copies of
// w_init / b_init so phase 2 has a branch-free K loop.
// ---------------------------------------------------------------------------
__global__ __launch_bounds__(1024) void bt_coef_kernel(
    const float* __restrict__ w_init, const float* __restrict__ b_init,
    const float* __restrict__ wv,
    float* __restrict__ c_pad, float* __restrict__ w_pad, float* __restrict__ b_pad)
{
  __shared__ float bufA[NP1];
  __shared__ float bufB[NP1];

  const float w0 = wv[0];
  const float w1 = wv[1];
  const int t = threadIdx.x;

  bufA[t] = (t == 0) ? 1.0f : 0.0f;
  if (t == 0) bufA[N_DIM] = 0.0f;
  __syncthreads();

  float* src = bufA;
  float* dst = bufB;
  for (int it = 0; it < N_DIM; ++it) {
    const float left = (t == 0) ? 0.0f : src[t - 1];
    dst[t] = w0 * src[t] + w1 * left;
    if (t == 0) dst[N_DIM] = w0 * src[N_DIM] + w1 * src[N_DIM - 1];
    __syncthreads();
    float* tmp = src; src = dst; dst = tmp;
  }

  // src holds the final coefficients. Emit padded arrays (tail -> 0).
  for (int j = t; j < KPAD; j += 1024) {
    const bool in = (j < NP1);
    c_pad[j] = in ? src[j]    : 0.0f;
    w_pad[j] = in ? w_init[j] : 0.0f;
    b_pad[j] = in ? b_init[j] : 0.0f;
  }
}

// ---------------------------------------------------------------------------
// Phase 2: out[b] = sum_j c[j] * relu(k[b]*w[j] + b[j])  as WMMA f32 GEMV.
// One wave per 16 batch rows. A-tile (16x4) built on the fly, B-tile (4x16)
// carries c in column 0 only (lane-masked multiply, no EXEC divergence in
// the loop -> EXEC stays all-1s across every v_wmma as the ISA requires).
//
// A 16x4 f32 layout: lanes 0-15 hold M=lane with {K=0,K=1} in VGPR{0,1};
//                    lanes 16-31 hold M=lane-16 with {K=2,K=3}.
// B 4x16 f32 layout: lanes 0-15 hold N=lane with {K=0,K=1};
//                    lanes 16-31 hold N=lane-16 with {K=2,K=3}.
// D column N=0 lives in lane 0 (M=0..7 in acc[0..7]) and lane 16 (M=8..15).
// ---------------------------------------------------------------------------
__global__ __launch_bounds__(256) void bt_gemv_wmma_kernel(
    const float* __restrict__ kvec, const float* __restrict__ c_pad,
    const float* __restrict__ w_pad, const float* __restrict__ b_pad,
    float* __restrict__ out)
{
  const int lane  = threadIdx.x & 31;
  const int wave  = blockIdx.x * (blockDim.x >> 5) + (threadIdx.x >> 5);
  const int row0  = wave * 16;
  const int m     = lane & 15;
  const float kk  = kvec[row0 + m];
  const float cmask = (m == 0) ? 1.0f : 0.0f;   // B column 0 only
  const int koff  = (lane < 16) ? 0 : 2;

  v8f acc = {};
  for (int k = 0; k < KPAD; k += 4) {
    const int j0 = k + koff;
    v2f A;
    A.x = fmaxf(kk * w_pad[j0]     + b_pad[j0],     0.0f);
    A.y = fmaxf(kk * w_pad[j0 + 1] + b_pad[j0 + 1], 0.0f);
    v2f B;
    B.x = c_pad[j0]     * cmask;
    B.y = c_pad[j0 + 1] * cmask;
    // 8 args: (neg_a, A, neg_b, B, c_mod, C, reuse_a, reuse_b)
    acc = __builtin_amdgcn_wmma_f32_16x16x4_f32(
        false, A, false, B, (short)0, acc, false, false);
  }

  if (m == 0) {
    const int base = row0 + (lane >> 4) * 8;   // lane 0 -> M=0..7, lane 16 -> M=8..15
#pragma unroll
    for (int i = 0; i < 8; ++i) out[base + i] = acc[i];
  }
}

// ---------------------------------------------------------------------------
// Launch. Inputs (setup_inputs order): k (B), w_init (1025), b_init (1025), w (2).
// Workspace: c_pad | w_pad | b_pad, KPAD floats each.
// ---------------------------------------------------------------------------
extern "C" void kernel_launch(void* const* d_in, const int* in_sizes, int n_in,
                              void* d_out, int out_size, void* d_ws, size_t ws_size,
                              hipStream_t stream) {
  (void)n_in; (void)out_size; (void)ws_size;
  const float* k      = (const float*)d_in[0];
  const float* w_init = (const float*)d_in[1];
  const float* b_init = (const float*)d_in[2];
  const float* wv     = (const float*)d_in[3];
  float* out = (float*)d_out;

  float* ws    = (float*)d_ws;
  float* c_pad = ws;
  float* w_pad = ws + KPAD;
  float* b_pad = ws + 2 * KPAD;

  const int batch = in_sizes[0];          // 8192
  bt_coef_kernel<<<1, 1024, 0, stream>>>(w_init, b_init, wv, c_pad, w_pad, b_pad);

  const int waves  = batch / 16;          // 512 wave-tiles
  const int blocks = (waves + 7) / 8;     // 8 waves (256 threads) per block
  bt_gemv_wmma_kernel<<<blocks, 256, 0, stream>>>(k, c_pad, w_pad, b_pad, out);
}